// LIF1Neuron_61409442398457
// MI455X (gfx1250) — compile-verified
//
#include <hip/hip_runtime.h>
#include <cstdint>
#include <cstddef>

// LIF neuron scan, MI455X (gfx1250).
// Memory-bound streaming kernel: 402 MB @ 23.3 TB/s => ~17us floor.
// Data path: global_load_async_to_lds_b128 (8-deep pipeline, ASYNCcnt) -> ds_load_b128
// -> VALU scan -> nontemporal global_store_b128.

typedef __attribute__((ext_vector_type(4))) float f4;

#define T_STEPS 32
#define NBUF    8          // async pipeline depth (LDS buffers)
#define BLOCK   256        // threads per block (8 waves of 32)
#define VEC     4          // floats per thread per time step (b128)
#define CHUNK   (BLOCK * VEC)   // 1024 floats per block per time slice

__device__ __forceinline__ void async_load_b128(uint32_t lds_addr, const void* gaddr) {
    // GV form: vdst = LDS byte address (VGPR), vaddr = 64-bit global address (VGPR pair)
    asm volatile("global_load_async_to_lds_b128 %0, %1, off"
                 :: "v"(lds_addr), "v"(gaddr)
                 : "memory");
}

__device__ __forceinline__ void wait_async_le(int n) {
    // n is a compile-time constant after full unroll; switch folds to one instruction.
    switch (n) {
    case 0: asm volatile("s_wait_asynccnt 0x0" ::: "memory"); break;
    case 1: asm volatile("s_wait_asynccnt 0x1" ::: "memory"); break;
    case 2: asm volatile("s_wait_asynccnt 0x2" ::: "memory"); break;
    case 3: asm volatile("s_wait_asynccnt 0x3" ::: "memory"); break;
    case 4: asm volatile("s_wait_asynccnt 0x4" ::: "memory"); break;
    case 5: asm volatile("s_wait_asynccnt 0x5" ::: "memory"); break;
    case 6: asm volatile("s_wait_asynccnt 0x6" ::: "memory"); break;
    default: asm volatile("s_wait_asynccnt 0x7" ::: "memory"); break;
    }
}

__device__ __forceinline__ void wait_ds0() {
    asm volatile("s_wait_dscnt 0x0" ::: "memory");
}

__global__ __launch_bounds__(BLOCK) void lif_scan_kernel(
        const float* __restrict__ x,   // (T, B*N) row-major
        float* __restrict__ out,       // spikes (T*M) then membranes (T*M)
        int M)                          // B*N floats per time slice
{
    __shared__ f4 smem[NBUF][BLOCK];   // 8 * 4KB = 32 KB; per-lane private slots

    const int tid   = threadIdx.x;
    const int gbase = blockIdx.x * CHUNK + tid * VEC;  // float index within a time slice
    const uint32_t lds0 = (uint32_t)(uintptr_t)&smem[0][tid];  // low 32 bits = LDS byte addr

    // ---- prologue: fill the 8-deep async pipeline ----
    #pragma unroll
    for (int s = 0; s < NBUF; ++s) {
        async_load_b128(lds0 + (uint32_t)s * (BLOCK * sizeof(f4)),
                        x + (size_t)s * M + gbase);
    }

    f4 m = {0.0f, 0.0f, 0.0f, 0.0f};   // membrane state (m0 = 0)

    float* spikes = out;
    float* membr  = out + (size_t)T_STEPS * M;

    #pragma unroll
    for (int t = 0; t < T_STEPS; ++t) {
        // Async loads complete in order: allow min(NBUF-1, T-1-t) still outstanding
        // => load for time t has landed in LDS.
        wait_async_le((T_STEPS - 1 - t) < (NBUF - 1) ? (T_STEPS - 1 - t) : (NBUF - 1));

        f4 v = smem[t % NBUF][tid];    // ds_load_b128 from this lane's slot
        wait_ds0();                    // v is in VGPRs; LDS slot is now free for reuse

        if (t + NBUF < T_STEPS) {      // refill the slot we just consumed
            async_load_b128(lds0 + (uint32_t)(t % NBUF) * (BLOCK * sizeof(f4)),
                            x + (size_t)(t + NBUF) * M + gbase);
        }

        // ---- LIF update: m = beta*m + x; spike = (m >= thr); record pre-reset m; soft reset ----
        f4 sp, mem;
        #pragma unroll
        for (int i = 0; i < VEC; ++i) {
            float mi = fmaf(0.25f, m[i], v[i]);     // BETA = 0.25
            float si = (mi >= 1.0f) ? 1.0f : 0.0f;  // THRESHOLD = 1.0
            mem[i] = mi;                            // membrane recorded BEFORE reset
            sp[i]  = si;
            m[i]   = mi - si;                       // soft reset (thr * spike)
        }

        const size_t o4 = ((size_t)t * M + gbase) >> 2;
        __builtin_nontemporal_store(sp,  ((f4*)spikes) + o4);  // streaming b128 NT stores
        __builtin_nontemporal_store(mem, ((f4*)membr)  + o4);
    }
}

extern "C" void kernel_launch(void* const* d_in, const int* in_sizes, int n_in,
                              void* d_out, int out_size, void* d_ws, size_t ws_size,
                              hipStream_t stream) {
    const float* x = (const float*)d_in[0];
    float* out = (float*)d_out;

    const int total = in_sizes[0];        // T * B * N = 33,554,432
    const int M = total / T_STEPS;        // B * N = 1,048,576 floats per time slice
    const int blocks = M / CHUNK;         // 1024 blocks

    lif_scan_kernel<<<blocks, BLOCK, 0, stream>>>(x, out, M);
}